// VQQuantizer_17892833755568
// MI455X (gfx1250) — compile-verified
//
#include <hip/hip_runtime.h>
#include <hip/hip_bf16.h>

typedef __attribute__((ext_vector_type(2))) float v2f;
typedef __attribute__((ext_vector_type(8))) float v8f;

#define N_TOT   65536      // B*H*W = 64*32*32
#define D_DIM   256
#define M_CODES 512
#define HW      1024       // H*W
#define ZQ_ELEMS 16777216  // 64*256*32*32

// ---------------------------------------------------------------------------
// Kernel 1: e_sq[m] = sum_d codebook[m][d]^2   (one wave per code)
// ---------------------------------------------------------------------------
__global__ __launch_bounds__(32) void vq_esq_kernel(const float* __restrict__ codebook,
                                                    float* __restrict__ esq) {
    const int code = blockIdx.x;
    const int lane = threadIdx.x;
    const float* row = codebook + (size_t)code * D_DIM;
    float s = 0.0f;
    #pragma unroll
    for (int j = 0; j < D_DIM / 32; ++j) {
        float v = row[lane + 32 * j];
        s += v * v;
    }
    #pragma unroll
    for (int off = 16; off >= 1; off >>= 1) s += __shfl_xor(s, off, 32);
    if (lane == 0) esq[code] = s;
}

// ---------------------------------------------------------------------------
// Kernel 2: WMMA fp32 distance GEMM + fused argmin.
// Block = 256 threads = 8 waves; wave w owns rows [blk*128 + w*16, +16).
// A-panel (16x256 of z_flat) resident in 64 v2f register fragments.
// Codebook streamed as 32x256 tiles through 32KB LDS; per tile each wave
// computes TWO 16x16 C tiles with FOUR independent WMMA accumulation
// chains (even/odd K), 128 x V_WMMA_F32_16X16X4_F32 per staged tile.
// ---------------------------------------------------------------------------
__global__ __launch_bounds__(256, 1) void vq_argmin_kernel(const float* __restrict__ z,
                                                           const float* __restrict__ codebook,
                                                           const float* __restrict__ esq,
                                                           float* __restrict__ idx_out_f) {
    __shared__ float sB[32 * D_DIM];   // 32 KB codebook tile, row-major [code][d]

    const int lane  = threadIdx.x & 31;
    const int wave  = threadIdx.x >> 5;
    const int lhalf = lane >> 4;       // 0: K=0,1 / rows 0..7 side, 1: K=2,3 / rows 8..15
    const int llow  = lane & 15;

    const int row_base = blockIdx.x * 128 + wave * 16;

    // --- load A panel: z_flat[row][d] = z[b*D*HW + d*HW + hw], n = b*HW + hw
    const int n  = row_base + llow;
    const int b  = n >> 10;
    const int hw = n & 1023;
    const float* zb = z + (size_t)b * (D_DIM * HW) + hw;

    v2f a[64];
    #pragma unroll
    for (int k4 = 0; k4 < 64; ++k4) {
        const int d0 = k4 * 4 + 2 * lhalf;       // ISA 16x4 f32 A layout
        v2f av;
        av.x = zb[(size_t)d0 * HW];
        av.y = zb[(size_t)(d0 + 1) * HW];
        a[k4] = av;
    }

    float best_s[8];
    int   best_i[8];
    #pragma unroll
    for (int v = 0; v < 8; ++v) { best_s[v] = 3.402823e38f; best_i[v] = 0; }

    #pragma unroll 1
    for (int mt = 0; mt < 16; ++mt) {
        const int m_base = mt * 32;

        __syncthreads();
        {   // cooperative stage of 32 codebook rows -> LDS (float4 x8 per thread)
            const int code = threadIdx.x >> 3;         // 0..31
            const int seg  = (threadIdx.x & 7) * 32;   // 32 consecutive floats
            const float4* src = (const float4*)(codebook + (size_t)(m_base + code) * D_DIM + seg);
            float4* dst = (float4*)(&sB[code * D_DIM + seg]);
            #pragma unroll
            for (int i = 0; i < 8; ++i) dst[i] = src[i];
        }
        __syncthreads();

        // two 16x16 fp32 accumulator tiles, each split into even/odd-K chains
        v8f c0a = {}, c0b = {}, c1a = {}, c1b = {};
        #pragma unroll
        for (int k4 = 0; k4 < 64; ++k4) {
            const int d0 = k4 * 4 + 2 * lhalf;   // B[k][ncol] = codebook[ncol][k]
            v2f b0 = *(const v2f*)(&sB[llow * D_DIM + d0]);         // codes m_base+llow
            v2f b1 = *(const v2f*)(&sB[(16 + llow) * D_DIM + d0]);  // codes m_base+16+llow
            if (k4 & 1) {
                c0b = __builtin_amdgcn_wmma_f32_16x16x4_f32(
                          false, a[k4], false, b0, (short)0, c0b, false, false);
                c1b = __builtin_amdgcn_wmma_f32_16x16x4_f32(
                          false, a[k4], false, b1, (short)0, c1b, false, false);
            } else {
                c0a = __builtin_amdgcn_wmma_f32_16x16x4_f32(
                          false, a[k4], false, b0, (short)0, c0a, false, false);
                c1a = __builtin_amdgcn_wmma_f32_16x16x4_f32(
                          false, a[k4], false, b1, (short)0, c1a, false, false);
            }
        }
        const v8f c0 = c0a + c0b;
        const v8f c1 = c1a + c1b;

        // score[n][m] = e_sq[m] - 2*dot ; update group 0 (lower m) first so
        // strict < keeps the first-occurrence argmin like jnp.argmin.
        const int   m0  = m_base + llow;
        const int   m1  = m_base + 16 + llow;
        const float es0 = esq[m0];
        const float es1 = esq[m1];
        #pragma unroll
        for (int v = 0; v < 8; ++v) {
            const float s0 = __builtin_fmaf(-2.0f, c0[v], es0);
            bool better = s0 < best_s[v];
            best_s[v] = better ? s0 : best_s[v];
            best_i[v] = better ? m0 : best_i[v];
            const float s1 = __builtin_fmaf(-2.0f, c1[v], es1);
            better = s1 < best_s[v];
            best_s[v] = better ? s1 : best_s[v];
            best_i[v] = better ? m1 : best_i[v];
        }
    }

    // argmin across the 16 lanes holding the same row (xor masks stay in-half)
    #pragma unroll
    for (int v = 0; v < 8; ++v) {
        #pragma unroll
        for (int off = 8; off >= 1; off >>= 1) {
            const float os = __shfl_xor(best_s[v], off, 32);
            const int   oi = __shfl_xor(best_i[v], off, 32);
            if (os < best_s[v] || (os == best_s[v] && oi < best_i[v])) {
                best_s[v] = os; best_i[v] = oi;
            }
        }
    }
    if (llow == 0) {
        #pragma unroll
        for (int v = 0; v < 8; ++v) {
            const int row = row_base + v + 8 * lhalf;
            idx_out_f[row] = (float)best_i[v];   // exact: idx < 512
        }
    }
}

// ---------------------------------------------------------------------------
// Kernel 3: z_q scatter (NHWC index -> NCHW layout) + per-block loss partials
// ---------------------------------------------------------------------------
__global__ __launch_bounds__(256) void vq_scatter_loss_kernel(const float* __restrict__ z,
                                                              const float* __restrict__ codebook,
                                                              const float* __restrict__ idx_f,
                                                              float* __restrict__ zq_out,
                                                              float* __restrict__ partials) {
    const int t = threadIdx.x;
    const int n = blockIdx.x * 256 + t;
    const int idx = (int)idx_f[n];
    const int b  = n >> 10;
    const int hw = n & 1023;
    const float* zb = z      + (size_t)b * (D_DIM * HW) + hw;
    float*       ob = zq_out + (size_t)b * (D_DIM * HW) + hw;
    const float* q  = codebook + (size_t)idx * D_DIM;

    float acc = 0.0f;
    #pragma unroll 8
    for (int d = 0; d < D_DIM; ++d) {
        const float qv = q[d];
        const float zv = zb[(size_t)d * HW];
        ob[(size_t)d * HW] = qv;                 // q_st == q numerically
        const float df = zv - qv;
        acc = __builtin_fmaf(df, df, acc);
    }

    __shared__ float red[8];
    #pragma unroll
    for (int off = 16; off >= 1; off >>= 1) acc += __shfl_xor(acc, off, 32);
    const int lane = t & 31, w = t >> 5;
    if (lane == 0) red[w] = acc;
    __syncthreads();
    if (w == 0) {
        float s = (lane < 8) ? red[lane] : 0.0f;
        #pragma unroll
        for (int off = 4; off >= 1; off >>= 1) s += __shfl_xor(s, off, 32);
        if (lane == 0) partials[blockIdx.x] = s;
    }
}

// ---------------------------------------------------------------------------
// Kernel 4: vq_loss = (1 + BETA) * sum / (N*D),  BETA = 0.25
// ---------------------------------------------------------------------------
__global__ __launch_bounds__(256) void vq_finalize_kernel(const float* __restrict__ partials,
                                                          float* __restrict__ loss_out) {
    float s = partials[threadIdx.x];
    __shared__ float red[8];
    const int lane = threadIdx.x & 31, w = threadIdx.x >> 5;
    #pragma unroll
    for (int off = 16; off >= 1; off >>= 1) s += __shfl_xor(s, off, 32);
    if (lane == 0) red[w] = s;
    __syncthreads();
    if (w == 0) {
        float v = (lane < 8) ? red[lane] : 0.0f;
        #pragma unroll
        for (int off = 4; off >= 1; off >>= 1) v += __shfl_xor(v, off, 32);
        if (lane == 0) *loss_out = v * (1.25f / 16777216.0f);
    }
}

// ---------------------------------------------------------------------------
extern "C" void kernel_launch(void* const* d_in, const int* in_sizes, int n_in,
                              void* d_out, int out_size, void* d_ws, size_t ws_size,
                              hipStream_t stream) {
    const float* z        = (const float*)d_in[0];   // (64,256,32,32) fp32
    const float* codebook = (const float*)d_in[1];   // (512,256) fp32

    float* out    = (float*)d_out;
    float* zq     = out;                 // 16,777,216 elements
    float* loss   = out + ZQ_ELEMS;      // 1 element
    float* idxf   = out + ZQ_ELEMS + 1;  // 65,536 elements (indices as float)

    float* esq      = (float*)d_ws;      // 512 floats
    float* partials = esq + M_CODES;     // 256 floats

    vq_esq_kernel<<<M_CODES, 32, 0, stream>>>(codebook, esq);
    vq_argmin_kernel<<<N_TOT / 128, 256, 0, stream>>>(z, codebook, esq, idxf);
    vq_scatter_loss_kernel<<<N_TOT / 256, 256, 0, stream>>>(z, codebook, idxf, zq, partials);
    vq_finalize_kernel<<<1, 256, 0, stream>>>(partials, loss);
}